// PhasedLSTM_18253611008554
// MI455X (gfx1250) — compile-verified
//
#include <hip/hip_runtime.h>
#include <math.h>

// Problem constants (B=64, T=512, H=768 from the reference)
#define HD   768
#define BD   64
#define TD   512
#define H4   3072            // 4*H
#define NBLK (HD / 16)       // 48 blocks, one 16-column slice each
#define KP   64              // K-panel staged in LDS
#define PSTR 68              // padded LDS row stride (floats) = 64 + TDM pad of 4 DWORDs
#define R_ON   0.05f
#define ALPHA_ 0.001f

typedef __attribute__((ext_vector_type(2))) float v2f;
typedef __attribute__((ext_vector_type(8))) float v8f;
typedef __attribute__((ext_vector_type(4))) unsigned int u32x4;
typedef __attribute__((ext_vector_type(8))) int i32x8;
typedef __attribute__((ext_vector_type(4))) int i32x4;

__device__ __forceinline__ float sigmoidf_(float v) { return 1.0f / (1.0f + expf(-v)); }

// ---------------------------------------------------------------------------
// TDM: DMA a 64x64 fp32 tile (row stride `row_stride_elems`) from global into
// LDS at byte offset `lds_off`, inserting 4 DWORDs of padding after every 64
// DWORDs => padded LDS row stride of 68 floats (matches PSTR).
//   D# group0: count=1 | lds_addr | global tile addr (57b) | type=2
//   D# group1: data_size=4B, pad_enable, pad_interval=5(64dw), pad_amount=3(4dw),
//              tensor_dim0/1 = 64, tile_dim0/1 = 64, tensor_dim0_stride = row stride
//   groups 2/3 (and trailing group): zero (2D tensor)
// ---------------------------------------------------------------------------
__device__ __forceinline__ void tdm_load_panel(unsigned lds_off,
                                               const float* __restrict__ gsrc,
                                               unsigned row_stride_elems) {
  unsigned long long ga = (unsigned long long)(uintptr_t)gsrc;
  u32x4 g0;
  g0[0] = 1u;                                            // count=1, no gather
  g0[1] = lds_off;                                       // lds_addr (bytes)
  g0[2] = (unsigned)(ga & 0xFFFFFFFFull);                // global_addr[31:0]
  g0[3] = (unsigned)((ga >> 32) & 0x01FFFFFFull)         // global_addr[56:32]
          | (2u << 30);                                  // type=2 ("image")
  i32x8 g1;
  g1[0] = (int)((2u << 16)                               // data_size: 4 bytes
                | (1u << 20)                             // pad_enable
                | (5u << 22)                             // pad_interval: 64 DWORDs
                | (3u << 25));                           // pad_amount: 4 DWORDs
  g1[1] = (int)(64u << 16);                              // tensor_dim0 = 64 (lo16)
  g1[2] = (int)(64u << 16);                              // dim0 hi=0 | tensor_dim1 lo = 64
  g1[3] = (int)(64u << 16);                              // dim1 hi=0 | tile_dim0 = 64
  g1[4] = (int)(64u);                                    // tile_dim1 = 64, tile_dim2 = 0
  g1[5] = (int)row_stride_elems;                         // tensor_dim0_stride[31:0]
  g1[6] = 0;                                             // stride0 hi | stride1 lo
  g1[7] = 0;                                             // stride1 hi
  i32x4 gz4 = {0, 0, 0, 0};
  i32x8 gz8 = {0, 0, 0, 0, 0, 0, 0, 0};
  // 6-arg toolchain form: (g0, g1, g2, g3, g4, cpol)
  __builtin_amdgcn_tensor_load_to_lds(g0, g1, gz4, gz4, gz8, 0);
}

// ---------------------------------------------------------------------------
// Prep: build transposed, gate-summed weight matrices.
//   UcatT[col*H + k] = column `col` of [Usum | Uc]  (col in [0,4H), k in [0,H))
// ---------------------------------------------------------------------------
__global__ void plstm_prep_weights(const float* __restrict__ W,
                                   const float* __restrict__ U,
                                   float* __restrict__ WcatT,
                                   float* __restrict__ UcatT) {
  int idx = blockIdx.x * blockDim.x + threadIdx.x;
  if (idx >= H4 * HD) return;
  int col = idx / HD;
  int k   = idx - col * HD;
  int g   = col / HD;            // gate: 0=i 1=f 2=o 3=c
  int j   = col - g * HD;
  float wu, uu;
  if (g < 3) {
    int c = g * HD + j;
    uu = U[(size_t)k * H4 + c] + U[(size_t)(HD + k) * H4 + c] + U[(size_t)(2 * HD + k) * H4 + c];
    wu = W[(size_t)k * H4 + c] + W[(size_t)(HD + k) * H4 + c] + W[(size_t)(2 * HD + k) * H4 + c];
  } else {
    int c = 3 * HD + j;
    uu = U[(size_t)(3 * HD + k) * H4 + c];
    wu = W[(size_t)(3 * HD + k) * H4 + c];
  }
  UcatT[idx] = uu;
  WcatT[idx] = wu;
}

// ---------------------------------------------------------------------------
// Prep: phase gate k[t, j]
// ---------------------------------------------------------------------------
__global__ void plstm_prep_kgate(const float* __restrict__ tau,
                                 const float* __restrict__ shift,
                                 float* __restrict__ kg) {
  int idx = blockIdx.x * blockDim.x + threadIdx.x;
  if (idx >= TD * HD) return;
  int t = idx / HD;
  int j = idx - t * HD;
  const float PI = 3.14159265358979323846f;
  float tj = tau[j];
  float a  = (float)t - shift[j];
  float fm = tj / PI * atanf(tanf(PI * (a / tj - 0.5f))) + 0.5f * tj;
  float phi = fm / (tj + 1e-8f);
  float k;
  if (phi < 0.5f * R_ON)      k = 2.0f * phi / R_ON;
  else if (phi < R_ON)        k = 2.0f - 2.0f * phi / R_ON;
  else                        k = ALPHA_ * phi;
  kg[idx] = k;
}

// ---------------------------------------------------------------------------
// Prep: zero h0/c0 (double buffered) and the grid-barrier counter
// ---------------------------------------------------------------------------
__global__ void plstm_init(float* __restrict__ hbuf, float* __restrict__ cbuf,
                           unsigned* __restrict__ cnt) {
  int idx = blockIdx.x * blockDim.x + threadIdx.x;
  if (idx < 2 * BD * HD) { hbuf[idx] = 0.0f; cbuf[idx] = 0.0f; }
  if (idx == 0) *cnt = 0u;
}

// ---------------------------------------------------------------------------
// Persistent recurrent kernel: 48 blocks x 128 threads (4 wave32).
// Block  b: hidden columns [16b, 16b+16).
// Wave   g: gate g of {i,f,o,c}; computes 4 M-tiles (64 batch rows) of
//           gates = h @ Ucat + x_t @ Wcat  via v_wmma_f32_16x16x4_f32.
// A-panels staged by the Tensor Data Mover (tensor_load_to_lds + TENSORcnt).
// One release/acquire grid barrier per timestep; h/c double-buffered.
// ---------------------------------------------------------------------------
__global__ __launch_bounds__(128, 1)
void plstm_persistent(const float* __restrict__ x,
                      const float* __restrict__ bias,
                      const float* __restrict__ wpeep,
                      const float* __restrict__ UcatT,
                      const float* __restrict__ WcatT,
                      const float* __restrict__ kgate,
                      float* __restrict__ hbuf,
                      float* __restrict__ cbuf,
                      float* __restrict__ out,
                      unsigned* __restrict__ cnt) {
  __shared__ float panel[KP * PSTR];    // K-panel of A (h or x_t), TDM-padded
  __shared__ float gl[4 * BD * 16];     // gate exchange: [gate][row 0..63][col 0..15]

  const int tid   = threadIdx.x;
  const int wave  = tid >> 5;           // gate index 0..3
  const int lane  = tid & 31;
  const int l15   = lane & 15;
  const int hi    = lane >> 4;          // half-wave select
  const int khalf = hi << 1;            // K base within 16x16x4 frag: 0 or 2
  const int hi8   = hi << 3;            // C/D row offset for lanes 16..31
  const int n0    = blockIdx.x << 4;    // hidden-column base

  // LDS byte offset of the staging panel (generic LDS addr low 32 bits)
  const unsigned panel_lds = (unsigned)(uintptr_t)&panel[0];

  // B-fragment source rows (columns of Ucat/Wcat, stored transposed)
  const float* Bu = UcatT + (size_t)(wave * HD + n0 + l15) * HD;
  const float* Bw = WcatT + (size_t)(wave * HD + n0 + l15) * HD;

  for (int t = 0; t < TD; ++t) {
    const float* hsrc = hbuf + (size_t)(t & 1) * (BD * HD);
    float*       hdst = hbuf + (size_t)((t + 1) & 1) * (BD * HD);
    const float* csrc = cbuf + (size_t)(t & 1) * (BD * HD);
    float*       cdst = cbuf + (size_t)((t + 1) & 1) * (BD * HD);

    v8f acc0 = {}, acc1 = {}, acc2 = {}, acc3 = {};

    for (int phase = 0; phase < 2; ++phase) {
      const float* Bt = phase ? Bw : Bu;
      for (int kp = 0; kp < HD; kp += KP) {
        __syncthreads();                        // previous panel fully consumed
        if (wave == 0) {
          // tile start = row 0 of the K-panel; rows are batch entries
          const float* src = phase ? (x + (size_t)t * HD + kp)   // row stride T*H
                                   : (hsrc + kp);                // row stride H
          tdm_load_panel(panel_lds, src,
                         phase ? (unsigned)(TD * HD) : (unsigned)HD);
          __builtin_amdgcn_s_wait_tensorcnt(0); // DMA complete -> LDS valid
        }
        __syncthreads();                        // release panel to all waves
        const float* brow = Bt + kp + khalf;
#pragma unroll 4
        for (int kk = 0; kk < KP; kk += 4) {
          v2f bf = *(const v2f*)(brow + kk);  // B frag: lane=N, {K=khalf, khalf+1}
          v2f a0 = *(const v2f*)&panel[( 0 + l15) * PSTR + kk + khalf];
          v2f a1 = *(const v2f*)&panel[(16 + l15) * PSTR + kk + khalf];
          v2f a2 = *(const v2f*)&panel[(32 + l15) * PSTR + kk + khalf];
          v2f a3 = *(const v2f*)&panel[(48 + l15) * PSTR + kk + khalf];
          acc0 = __builtin_amdgcn_wmma_f32_16x16x4_f32(false, a0, false, bf, (short)0, acc0, false, false);
          acc1 = __builtin_amdgcn_wmma_f32_16x16x4_f32(false, a1, false, bf, (short)0, acc1, false, false);
          acc2 = __builtin_amdgcn_wmma_f32_16x16x4_f32(false, a2, false, bf, (short)0, acc2, false, false);
          acc3 = __builtin_amdgcn_wmma_f32_16x16x4_f32(false, a3, false, bf, (short)0, acc3, false, false);
        }
      }
    }

    // scatter accumulator tiles to LDS: gl[gate][row][col]
    {
      const int base = wave * 1024 + l15;
#pragma unroll
      for (int v = 0; v < 8; ++v) {
        gl[base + ( 0 + v + hi8) * 16] = acc0[v];
        gl[base + (16 + v + hi8) * 16] = acc1[v];
        gl[base + (32 + v + hi8) * 16] = acc2[v];
        gl[base + (48 + v + hi8) * 16] = acc3[v];
      }
    }
    __syncthreads();

    // fused elementwise: bias + peephole + sigmoid/tanh + phase gate
#pragma unroll
    for (int r = 0; r < 8; ++r) {
      int idx = tid + (r << 7);       // 0..1023
      int row = idx >> 4;             // batch index
      int col = idx & 15;
      int j   = n0 + col;
      size_t gidx = (size_t)row * HD + j;
      float cp = csrc[gidx];
      float hp = hsrc[gidx];
      float gi = gl[0 * 1024 + idx] + bias[j]          + wpeep[j]          * cp;
      float gf = gl[1 * 1024 + idx] + bias[HD + j]     + wpeep[HD + j]     * cp;
      float go = gl[2 * 1024 + idx] + bias[2 * HD + j] + wpeep[2 * HD + j] * cp;
      float gc = gl[3 * 1024 + idx] + bias[3 * HD + j];
      float it = sigmoidf_(gi);
      float ft = sigmoidf_(gf);
      float ot = sigmoidf_(go);
      float cprim = ft * cp + it * tanhf(gc);
      float kt = kgate[(size_t)t * HD + j];
      float cnew = kt * cprim + (1.0f - kt) * cp;
      float hnew = kt * (tanhf(cprim) * ot) + (1.0f - kt) * hp;
      cdst[gidx] = cnew;
      hdst[gidx] = hnew;
      out[((size_t)row * TD + t) * HD + j] = hnew;
    }

    // grid barrier between timesteps (skip after the last one)
    if (t + 1 < TD) {
      __threadfence();                 // push h/c stores device-visible
      __syncthreads();
      const unsigned target = (unsigned)gridDim.x * (unsigned)(t + 1);
      if (tid == 0)
        __hip_atomic_fetch_add(cnt, 1u, __ATOMIC_RELEASE, __HIP_MEMORY_SCOPE_AGENT);
      if (lane == 0) {                 // one spinner per wave -> acquire on every CU
        while (__hip_atomic_load(cnt, __ATOMIC_ACQUIRE, __HIP_MEMORY_SCOPE_AGENT) < target)
          __builtin_amdgcn_s_sleep(2);
      }
      __syncthreads();
    }
  }
}

// ---------------------------------------------------------------------------
// Host launcher
// ---------------------------------------------------------------------------
extern "C" void kernel_launch(void* const* d_in, const int* in_sizes, int n_in,
                              void* d_out, int out_size, void* d_ws, size_t ws_size,
                              hipStream_t stream) {
  (void)in_sizes; (void)n_in; (void)out_size; (void)ws_size;
  const float* x     = (const float*)d_in[0];  // (B,T,H)
  const float* W     = (const float*)d_in[1];  // (4H,4H)
  const float* U     = (const float*)d_in[2];  // (4H,4H)
  const float* b     = (const float*)d_in[3];  // (4H,)
  const float* wpeep = (const float*)d_in[4];  // (3H,)
  const float* tau   = (const float*)d_in[5];  // (H,)
  const float* shift = (const float*)d_in[6];  // (H,)
  float* out = (float*)d_out;                  // (B,T,H)

  // workspace layout (floats)
  float* ws    = (float*)d_ws;
  float* UcatT = ws;                           // 4H*H
  float* WcatT = UcatT + (size_t)H4 * HD;      // 4H*H
  float* kg    = WcatT + (size_t)H4 * HD;      // T*H
  float* hbuf  = kg + (size_t)TD * HD;         // 2*B*H
  float* cbuf  = hbuf + 2 * BD * HD;           // 2*B*H
  unsigned* cnt = (unsigned*)(cbuf + 2 * BD * HD);

  {
    int n = H4 * HD;
    plstm_prep_weights<<<(n + 255) / 256, 256, 0, stream>>>(W, U, WcatT, UcatT);
  }
  {
    int n = TD * HD;
    plstm_prep_kgate<<<(n + 255) / 256, 256, 0, stream>>>(tau, shift, kg);
  }
  {
    int n = 2 * BD * HD;
    plstm_init<<<(n + 255) / 256, 256, 0, stream>>>(hbuf, cbuf, cnt);
  }
  plstm_persistent<<<NBLK, 128, 0, stream>>>(x, b, wpeep, UcatT, WcatT, kg,
                                             hbuf, cbuf, out, cnt);
}